// spline_max_pool_34677565948744
// MI455X (gfx1250) — compile-verified
//
#include <hip/hip_runtime.h>
#include <math.h>

#define THREADS 256

typedef __attribute__((ext_vector_type(2))) float f32x2;
typedef __attribute__((ext_vector_type(8))) float f32x8;

// ---------- helpers ----------
static __device__ __forceinline__ unsigned ford(float f) {
  unsigned u = __float_as_uint(f);
  return (u >> 31) ? ~u : (u | 0x80000000u);   // order-preserving map for atomicMax
}
static __device__ __forceinline__ float funord(unsigned u) {
  return (u >> 31) ? __uint_as_float(u ^ 0x80000000u) : __uint_as_float(~u);
}
static __device__ __forceinline__ float eluf(float v) {
  return v > 0.0f ? v : (expf(v) - 1.0f);
}

// ---------- tiny setup ----------
__global__ void k_init_misc(int* ecnt, int e0) {
  if (threadIdx.x == 0) ecnt[0] = e0;
}

__global__ void k_convert_edges(const long long* __restrict__ eidx, int* __restrict__ s,
                                int* __restrict__ d, float* __restrict__ deg, int e0) {
  int e = blockIdx.x * blockDim.x + threadIdx.x;
  if (e >= e0) return;
  int si = (int)eidx[e];
  int di = (int)eidx[e0 + e];
  s[e] = si; d[e] = di;
  atomicAdd(&deg[di], 1.0f);
}

// ---------- edge pseudo-coordinate machinery ----------
__global__ void k_edge_absmax(const float* __restrict__ pos, const int* __restrict__ src,
                              const int* __restrict__ dst, const int* __restrict__ cnt,
                              unsigned* __restrict__ mslot) {
  int e = blockIdx.x * blockDim.x + threadIdx.x;
  if (e >= *cnt) return;
  int s = src[e], d = dst[e];
  float ax = fabsf(pos[s * 3 + 0] - pos[d * 3 + 0]);
  float ay = fabsf(pos[s * 3 + 1] - pos[d * 3 + 1]);
  float az = fabsf(pos[s * 3 + 2] - pos[d * 3 + 2]);
  float a = fmaxf(ax, fmaxf(ay, az));          // >= 0 -> raw bits are monotone
  atomicMax(mslot, __float_as_uint(a));
}

__global__ void k_edge_basis(const float* __restrict__ pos, const int* __restrict__ src,
                             const int* __restrict__ dst, const int* __restrict__ cnt,
                             const unsigned* __restrict__ mslot, float* __restrict__ basis) {
  int e = blockIdx.x * blockDim.x + threadIdx.x;
  if (e >= *cnt) return;
  float m = fmaxf(__uint_as_float(*mslot), 1e-9f);
  float inv = 1.0f / (2.0f * m);
  int s = src[e], d = dst[e];
  float u0 = (pos[s * 3 + 0] - pos[d * 3 + 0]) * inv + 0.5f;
  float u1 = (pos[s * 3 + 1] - pos[d * 3 + 1]) * inv + 0.5f;
  float u2 = (pos[s * 3 + 2] - pos[d * 3 + 2]) * inv + 0.5f;
  #pragma unroll
  for (int k = 0; k < 8; ++k) {
    float w = ((k & 4) ? u0 : 1.0f - u0) * ((k & 2) ? u1 : 1.0f - u1) * ((k & 1) ? u2 : 1.0f - u2);
    basis[e * 8 + k] = w;
  }
}

// t[dst, k*ci + c] += basis[e,k] * x[src, c]
__global__ void k_scatter_t(const float* __restrict__ x, const int* __restrict__ src,
                            const int* __restrict__ dst, const float* __restrict__ basis,
                            const int* __restrict__ cnt, float* __restrict__ t,
                            int ci, int total) {
  int idx = blockIdx.x * blockDim.x + threadIdx.x;
  if (idx >= total) return;
  int e = idx / ci, c = idx - e * ci;
  if (e >= *cnt) return;
  float xv = x[(size_t)src[e] * ci + c];
  const float* bs = basis + (size_t)e * 8;
  float* td = t + (size_t)dst[e] * ci * 8 + c;
  #pragma unroll
  for (int k = 0; k < 8; ++k) atomicAdd(td + (size_t)k * ci, bs[k] * xv);
}

__global__ void k_scale_t(float* __restrict__ t, const float* __restrict__ deg,
                          int rowlen, int total) {
  int idx = blockIdx.x * blockDim.x + threadIdx.x;
  if (idx >= total) return;
  t[idx] *= 1.0f / fmaxf(deg[idx / rowlen], 1.0f);
}

__global__ void k_init_bias(float* __restrict__ y, const float* __restrict__ b, int co, int total) {
  int idx = blockIdx.x * blockDim.x + threadIdx.x;
  if (idx >= total) return;
  y[idx] = b[idx % co];
}

__global__ void k_y1_init(const float* __restrict__ x, const float* __restrict__ Wr1,
                          const float* __restrict__ b1, float* __restrict__ y, int co, int total) {
  int idx = blockIdx.x * blockDim.x + threadIdx.x;
  if (idx >= total) return;
  int n = idx / co, c = idx - n * co;
  y[idx] = x[n] * Wr1[c] + b1[c];
}

__global__ void k_elu(float* __restrict__ p, int n) {
  int i = blockIdx.x * blockDim.x + threadIdx.x;
  if (i < n) p[i] = eluf(p[i]);
}

// ---------- WMMA f32 GEMM: C += A[MxK] @ B[KxN], C preloaded ----------
// One wave computes a 16x64 tile (4 accumulators share one A fragment).
// OOB M rows are clamped at load (their results are garbage but never stored).
__global__ __launch_bounds__(THREADS) void k_gemm_f32wmma(
    const float* __restrict__ A, const float* __restrict__ Bw, float* __restrict__ C,
    int M, int K, int N) {
  int wave = blockIdx.x * (blockDim.x >> 5) + (threadIdx.x >> 5);
  int lane = threadIdx.x & 31;
  int ntiles = N >> 6;                    // 64-wide tiles
  int mtiles = (M + 15) >> 4;
  if (wave >= ntiles * mtiles) return;    // wave-uniform exit, EXEC stays full
  int mt = wave / ntiles;
  int nt = wave - mt * ntiles;
  int half = lane >> 4;                   // K-half select per ISA 16x4 layout
  int l16 = lane & 15;
  int arow = mt * 16 + l16;
  if (arow >= M) arow = M - 1;            // clamp: safe address, result discarded
  int koffb = half * 2;
  int ncol = nt * 64 + l16;
  const f32x2* ap = (const f32x2*)(A + (size_t)arow * K + koffb);
  const float* bp = Bw + (size_t)koffb * N + ncol;
  f32x8 acc0, acc1, acc2, acc3;
  #pragma unroll
  for (int v = 0; v < 8; ++v) {
    int row = mt * 16 + v + half * 8;
    int rc = row < M ? row : M - 1;
    const float* cr = C + (size_t)rc * N + ncol;
    acc0[v] = cr[0];
    acc1[v] = cr[16];
    acc2[v] = cr[32];
    acc3[v] = cr[48];
  }
  #pragma unroll 2
  for (int kk = 0; kk < K; kk += 4) {
    f32x2 af = *ap; ap += 2;              // 4 floats per iter along K
    f32x2 bf0, bf1, bf2, bf3;
    bf0.x = bp[0];   bf0.y = bp[N];
    bf1.x = bp[16];  bf1.y = bp[N + 16];
    bf2.x = bp[32];  bf2.y = bp[N + 32];
    bf3.x = bp[48];  bf3.y = bp[N + 48];
    bp += (size_t)4 * N;
    acc0 = __builtin_amdgcn_wmma_f32_16x16x4_f32(false, af, false, bf0,
                                                 (short)0, acc0, false, false);
    acc1 = __builtin_amdgcn_wmma_f32_16x16x4_f32(false, af, false, bf1,
                                                 (short)0, acc1, false, false);
    acc2 = __builtin_amdgcn_wmma_f32_16x16x4_f32(false, af, false, bf2,
                                                 (short)0, acc2, false, false);
    acc3 = __builtin_amdgcn_wmma_f32_16x16x4_f32(false, af, false, bf3,
                                                 (short)0, acc3, false, false);
  }
  #pragma unroll
  for (int v = 0; v < 8; ++v) {
    int row = mt * 16 + v + half * 8;
    if (row < M) {
      float* cr = C + (size_t)row * N + ncol;
      cr[0]  = acc0[v];
      cr[16] = acc1[v];
      cr[32] = acc2[v];
      cr[48] = acc3[v];
    }
  }
}

// split-K variant for skinny-M (FC1): 16x64 tile per wave, partials atomically reduced
__global__ __launch_bounds__(THREADS) void k_gemm_splitk_f32wmma(
    const float* __restrict__ A, const float* __restrict__ Bw, float* __restrict__ C,
    int M, int K, int N, int kchunks) {
  int wave = blockIdx.x * (blockDim.x >> 5) + (threadIdx.x >> 5);
  int lane = threadIdx.x & 31;
  int ntiles = N >> 6;
  if (wave >= ntiles * kchunks) return;
  int nt = wave / kchunks;
  int kc = wave - nt * kchunks;
  int klen = K / kchunks, k0 = kc * klen;
  int half = lane >> 4;
  int l16 = lane & 15;
  int koffb = half * 2;
  int arow = l16 < M ? l16 : M - 1;       // clamp: rows >= M never stored
  int ncol = nt * 64 + l16;
  const f32x2* ap = (const f32x2*)(A + (size_t)arow * K + k0 + koffb);
  const float* bp = Bw + (size_t)(k0 + koffb) * N + ncol;
  f32x8 acc0 = {}, acc1 = {}, acc2 = {}, acc3 = {};
  #pragma unroll 2
  for (int kk = 0; kk < klen; kk += 4) {
    f32x2 af = *ap; ap += 2;
    f32x2 bf0, bf1, bf2, bf3;
    bf0.x = bp[0];   bf0.y = bp[N];
    bf1.x = bp[16];  bf1.y = bp[N + 16];
    bf2.x = bp[32];  bf2.y = bp[N + 32];
    bf3.x = bp[48];  bf3.y = bp[N + 48];
    if (kk + 32 < klen) __builtin_prefetch(bp + (size_t)32 * N, 0, 1);
    bp += (size_t)4 * N;
    acc0 = __builtin_amdgcn_wmma_f32_16x16x4_f32(false, af, false, bf0,
                                                 (short)0, acc0, false, false);
    acc1 = __builtin_amdgcn_wmma_f32_16x16x4_f32(false, af, false, bf1,
                                                 (short)0, acc1, false, false);
    acc2 = __builtin_amdgcn_wmma_f32_16x16x4_f32(false, af, false, bf2,
                                                 (short)0, acc2, false, false);
    acc3 = __builtin_amdgcn_wmma_f32_16x16x4_f32(false, af, false, bf3,
                                                 (short)0, acc3, false, false);
  }
  #pragma unroll
  for (int v = 0; v < 8; ++v) {
    int row = v + half * 8;
    if (row < M) {
      float* cr = C + (size_t)row * N + ncol;
      atomicAdd(&cr[0],  acc0[v]);
      atomicAdd(&cr[16], acc1[v]);
      atomicAdd(&cr[32], acc2[v]);
      atomicAdd(&cr[48], acc3[v]);
    }
  }
}

// ---------- voxel max-pool ----------
__global__ void k_fill_ordneg(unsigned* __restrict__ p, int n) {
  int i = blockIdx.x * blockDim.x + threadIdx.x;
  if (i < n) p[i] = ford(-1e30f);
}

__global__ void k_pool_node(const float* __restrict__ posprev, const float* __restrict__ cntprev,
                            int* __restrict__ nid, float* __restrict__ cntnew,
                            float* __restrict__ posnew, int nprev, int perGraph,
                            float vs, int G, int cells) {
  int n = blockIdx.x * blockDim.x + threadIdx.x;
  if (n >= nprev) return;
  float px = posprev[n * 3 + 0], py = posprev[n * 3 + 1], pz = posprev[n * 3 + 2];
  int cx = (int)floorf(px / vs); cx = cx < 0 ? 0 : (cx > G - 1 ? G - 1 : cx);
  int cy = (int)floorf(py / vs); cy = cy < 0 ? 0 : (cy > G - 1 ? G - 1 : cy);
  int cz = (int)floorf(pz / vs); cz = cz < 0 ? 0 : (cz > G - 1 ? G - 1 : cz);
  int id = (n / perGraph) * cells + (cx * G + cy) * G + cz;
  nid[n] = id;
  bool valid = cntprev ? (cntprev[n] > 0.0f) : true;
  if (!valid) return;
  atomicAdd(&cntnew[id], 1.0f);
  atomicAdd(&posnew[id * 3 + 0], px);
  atomicAdd(&posnew[id * 3 + 1], py);
  atomicAdd(&posnew[id * 3 + 2], pz);
}

__global__ void k_pool_feat(const float* __restrict__ y, const int* __restrict__ nid,
                            const float* __restrict__ cntprev, unsigned* __restrict__ xpu,
                            int co, int total) {
  int idx = blockIdx.x * blockDim.x + threadIdx.x;
  if (idx >= total) return;
  int n = idx / co, c = idx - n * co;
  if (cntprev && !(cntprev[n] > 0.0f)) return;
  atomicMax(&xpu[(size_t)nid[n] * co + c], ford(y[idx]));
}

__global__ void k_pool_fin(unsigned* __restrict__ xpu, const float* __restrict__ cnt,
                           float* __restrict__ posp, int co, int total) {
  int idx = blockIdx.x * blockDim.x + threadIdx.x;
  if (idx >= total) return;
  int i = idx / co, c = idx - i * co;
  float ct = cnt[i];
  float v = funord(xpu[idx]);
  ((float*)xpu)[idx] = (ct > 0.0f) ? v : 0.0f;
  if (c < 3) posp[i * 3 + c] /= fmaxf(ct, 1.0f);
}

// unique-ify coarsened edges via bitmap claim + compaction
__global__ void k_coarsen(const int* __restrict__ src, const int* __restrict__ dst,
                          const int* __restrict__ cnt, const int* __restrict__ nid,
                          unsigned* __restrict__ bitmap, int* __restrict__ nsrc,
                          int* __restrict__ ndst, int* __restrict__ ncnt,
                          float* __restrict__ ndeg, int nnew) {
  int e = blockIdx.x * blockDim.x + threadIdx.x;
  if (e >= *cnt) return;
  int s2 = nid[src[e]], d2 = nid[dst[e]];
  if (s2 == d2) return;
  long long key = (long long)s2 * nnew + d2;
  unsigned bit = 1u << (key & 31);
  unsigned old = atomicOr(&bitmap[key >> 5], bit);
  if (old & bit) return;
  int i = atomicAdd(ncnt, 1);
  nsrc[i] = s2; ndst[i] = d2;
  atomicAdd(&ndeg[d2], 1.0f);
}

// ---------- final MaxPoolingX + FC ----------
__global__ void k_finalpool(const float* __restrict__ y4, const float* __restrict__ pos3,
                            const float* __restrict__ cnt3, unsigned* __restrict__ xgu,
                            int total) {
  int idx = blockIdx.x * blockDim.x + threadIdx.x;
  if (idx >= total) return;
  int n = idx / 512, c = idx - n * 512;
  if (!(cnt3[n] > 0.0f)) return;
  float px = pos3[n * 3 + 0], py = pos3[n * 3 + 1], pz = pos3[n * 3 + 2];
  int cx = (int)floorf(px / 80.0f); cx = cx < 0 ? 0 : (cx > 2 ? 2 : cx);
  int cy = (int)floorf(py / 80.0f); cy = cy < 0 ? 0 : (cy > 2 ? 2 : cy);
  int cz = (int)floorf(pz / 80.0f); cz = cz < 0 ? 0 : (cz > 2 ? 2 : cz);
  int cluster = (n / 125) * 64 + (cx * 3 + cy) * 3 + cz;
  atomicMax(&xgu[(size_t)cluster * 512 + c], ford(y4[idx]));
}

__global__ void k_xg_fin(unsigned* __restrict__ xgu, int total) {
  int idx = blockIdx.x * blockDim.x + threadIdx.x;
  if (idx >= total) return;
  float v = funord(xgu[idx]);
  ((float*)xgu)[idx] = (v <= -5e29f) ? 0.0f : v;
}

__global__ void k_fc2(const float* __restrict__ h, const float* __restrict__ w,
                      const float* __restrict__ b, float* __restrict__ out) {
  __shared__ float red[THREADS];
  int bi = blockIdx.x;
  float s = 0.0f;
  for (int i = threadIdx.x; i < 1024; i += THREADS) s += h[bi * 1024 + i] * w[i];
  red[threadIdx.x] = s;
  __syncthreads();
  for (int st = THREADS / 2; st > 0; st >>= 1) {
    if ((int)threadIdx.x < st) red[threadIdx.x] += red[threadIdx.x + st];
    __syncthreads();
  }
  if (threadIdx.x == 0) out[bi] = eluf(red[0] + b[0]);
}

// ---------- host ----------
extern "C" void kernel_launch(void* const* d_in, const int* in_sizes, int n_in,
                              void* d_out, int out_size, void* d_ws, size_t ws_size,
                              hipStream_t stream) {
  (void)in_sizes; (void)n_in; (void)out_size; (void)ws_size;
  constexpr int N0 = 20000, E0 = 160000;
  constexpr int N1 = 4 * 1728, N2 = 4 * 512, N3 = 4 * 125;     // 6912, 2048, 500
  constexpr int C1 = 64, C2 = 128, C3 = 256, C4 = 512;
  constexpr size_t W1BITS = (size_t)N1 * N1, W2BITS = (size_t)N2 * N2, W3BITS = (size_t)N3 * N3;
  const size_t words1 = (W1BITS + 31) / 32, words2 = (W2BITS + 31) / 32, words3 = (W3BITS + 31) / 32;

  const float*     x     = (const float*)d_in[0];
  const float*     pos   = (const float*)d_in[1];
  const long long* eidx  = (const long long*)d_in[2];
  const float *W1 = (const float*)d_in[4],  *Wr1 = (const float*)d_in[5],  *b1 = (const float*)d_in[6];
  const float *W2 = (const float*)d_in[7],  *Wr2 = (const float*)d_in[8],  *b2 = (const float*)d_in[9];
  const float *W3 = (const float*)d_in[10], *Wr3 = (const float*)d_in[11], *b3 = (const float*)d_in[12];
  const float *W4 = (const float*)d_in[13], *Wr4 = (const float*)d_in[14], *b4 = (const float*)d_in[15];
  const float *fc1w = (const float*)d_in[16], *fc1b = (const float*)d_in[17];
  const float *fc2w = (const float*)d_in[18], *fc2b = (const float*)d_in[19];
  float* out = (float*)d_out;

  char* Wb = (char*)d_ws;
  size_t off = 0;
  auto take = [&](size_t bytes) { char* p = Wb + off; off = (off + bytes + 255) & ~(size_t)255; return p; };
  int* src0 = (int*)take((size_t)E0 * 4);   int* dst0 = (int*)take((size_t)E0 * 4);
  int* e1s  = (int*)take((size_t)E0 * 4);   int* e1d  = (int*)take((size_t)E0 * 4);
  int* e2s  = (int*)take((size_t)E0 * 4);   int* e2d  = (int*)take((size_t)E0 * 4);
  int* e3s  = (int*)take((size_t)E0 * 4);   int* e3d  = (int*)take((size_t)E0 * 4);
  int* misc = (int*)take(64 * 4);
  int* ecnt = misc;                  // [0..3] edge counts per stage
  unsigned* mm = (unsigned*)(misc + 8);  // [0..3] pseudo-max per layer
  float* deg0 = (float*)take((size_t)N0 * 4);
  float* deg1 = (float*)take((size_t)N1 * 4);
  float* deg2 = (float*)take((size_t)N2 * 4);
  float* deg3 = (float*)take((size_t)N3 * 4);
  float* basis = (float*)take((size_t)E0 * 8 * 4);
  float* tbuf  = (float*)take((size_t)N1 * 8 * C1 * 4 > (size_t)N2 * 8 * C2 * 4 ?
                              (size_t)N1 * 8 * C1 * 4 : (size_t)N2 * 8 * C2 * 4); // 14.2MB max (N1*512)
  float* y1 = (float*)take((size_t)N0 * C1 * 4);
  unsigned* x1u = (unsigned*)take((size_t)N1 * C1 * 4);
  float* pos1 = (float*)take((size_t)N1 * 12);  float* cnt1 = (float*)take((size_t)N1 * 4);
  int* nid0 = (int*)take((size_t)N0 * 4);
  float* y2 = (float*)take((size_t)N1 * C2 * 4);
  unsigned* x2u = (unsigned*)take((size_t)N2 * C2 * 4);
  float* pos2 = (float*)take((size_t)N2 * 12);  float* cnt2 = (float*)take((size_t)N2 * 4);
  int* nid1 = (int*)take((size_t)N1 * 4);
  float* y3 = (float*)take((size_t)N2 * C3 * 4);
  unsigned* x3u = (unsigned*)take((size_t)N3 * C3 * 4);
  float* pos3 = (float*)take((size_t)N3 * 12);  float* cnt3 = (float*)take((size_t)N3 * 4);
  int* nid2 = (int*)take((size_t)N2 * 4);
  float* y4 = (float*)take((size_t)N3 * C4 * 4);
  unsigned* xgu = (unsigned*)take((size_t)256 * 512 * 4);
  float* h = (float*)take((size_t)4 * 1024 * 4);
  unsigned* bitmap = (unsigned*)take(words1 * 4);

  auto nb = [](long long n) { return (unsigned)((n + THREADS - 1) / THREADS); };
  const unsigned EB = nb(E0);
  auto gemm = [&](const float* A, const float* Bm, float* C, int M, int K, int N) {
    int tiles = ((M + 15) / 16) * (N / 64);
    k_gemm_f32wmma<<<dim3((tiles + 7) / 8), dim3(THREADS), 0, stream>>>(A, Bm, C, M, K, N);
  };
  auto conv = [&](const float* xin, const float* posin, int* es, int* ed, int* cp, float* deg,
                  const float* Wk, const float* Wr, const float* bias, float* y,
                  int n, int ci, int co, unsigned* ms) {
    k_edge_absmax<<<EB, THREADS, 0, stream>>>(posin, es, ed, cp, ms);
    k_edge_basis<<<EB, THREADS, 0, stream>>>(posin, es, ed, cp, ms, basis);
    hipMemsetAsync(tbuf, 0, (size_t)n * 8 * ci * 4, stream);
    long long sc = (long long)E0 * ci;
    k_scatter_t<<<nb(sc), THREADS, 0, stream>>>(xin, es, ed, basis, cp, tbuf, ci, (int)sc);
    k_scale_t<<<nb((long long)n * 8 * ci), THREADS, 0, stream>>>(tbuf, deg, 8 * ci, n * 8 * ci);
    k_init_bias<<<nb((long long)n * co), THREADS, 0, stream>>>(y, bias, co, n * co);
    gemm(tbuf, Wk, y, n, 8 * ci, co);
    gemm(xin, Wr, y, n, ci, co);
    k_elu<<<nb((long long)n * co), THREADS, 0, stream>>>(y, n * co);
  };
  auto pool = [&](const float* yin, const float* posin, const float* cntprev, int nprev,
                  int perGraph, float vs, int G, int cells, int co, int* nidprev,
                  float* cntnew, float* posnew, unsigned* xpu,
                  int* eso, int* edo, int* cpo, int* esn, int* edn, int* cpn,
                  float* degn, size_t bwords) {
    int ncellsTot = 4 * cells;
    k_fill_ordneg<<<nb((long long)ncellsTot * co), THREADS, 0, stream>>>(xpu, ncellsTot * co);
    hipMemsetAsync(cntnew, 0, (size_t)ncellsTot * 4, stream);
    hipMemsetAsync(posnew, 0, (size_t)ncellsTot * 12, stream);
    k_pool_node<<<nb(nprev), THREADS, 0, stream>>>(posin, cntprev, nidprev, cntnew, posnew,
                                                   nprev, perGraph, vs, G, cells);
    k_pool_feat<<<nb((long long)nprev * co), THREADS, 0, stream>>>(yin, nidprev, cntprev, xpu,
                                                                   co, nprev * co);
    k_pool_fin<<<nb((long long)ncellsTot * co), THREADS, 0, stream>>>(xpu, cntnew, posnew, co,
                                                                      ncellsTot * co);
    hipMemsetAsync(bitmap, 0, bwords * 4, stream);
    k_coarsen<<<EB, THREADS, 0, stream>>>(eso, edo, cpo, nidprev, bitmap, esn, edn, cpn,
                                          degn, ncellsTot);
  };

  // ---- setup ----
  hipMemsetAsync(misc, 0, 64 * 4, stream);
  k_init_misc<<<1, 64, 0, stream>>>(ecnt, E0);
  hipMemsetAsync(deg0, 0, (size_t)N0 * 4, stream);
  hipMemsetAsync(deg1, 0, (size_t)N1 * 4, stream);
  hipMemsetAsync(deg2, 0, (size_t)N2 * 4, stream);
  hipMemsetAsync(deg3, 0, (size_t)N3 * 4, stream);
  k_convert_edges<<<EB, THREADS, 0, stream>>>(eidx, src0, dst0, deg0, E0);

  // ---- conv1 (ci=1 special: y = x*Wr1 + b1, then += t@W1) ----
  k_edge_absmax<<<EB, THREADS, 0, stream>>>(pos, src0, dst0, ecnt + 0, mm + 0);
  k_edge_basis<<<EB, THREADS, 0, stream>>>(pos, src0, dst0, ecnt + 0, mm + 0, basis);
  hipMemsetAsync(tbuf, 0, (size_t)N0 * 8 * 4, stream);
  k_scatter_t<<<EB, THREADS, 0, stream>>>(x, src0, dst0, basis, ecnt + 0, tbuf, 1, E0);
  k_scale_t<<<nb((long long)N0 * 8), THREADS, 0, stream>>>(tbuf, deg0, 8, N0 * 8);
  k_y1_init<<<nb((long long)N0 * C1), THREADS, 0, stream>>>(x, Wr1, b1, y1, C1, N0 * C1);
  gemm(tbuf, W1, y1, N0, 8, C1);
  k_elu<<<nb((long long)N0 * C1), THREADS, 0, stream>>>(y1, N0 * C1);

  // ---- pool1 / conv2 ----
  pool(y1, pos, nullptr, N0, 5000, 20.0f, 12, 1728, C1, nid0, cnt1, pos1, x1u,
       src0, dst0, ecnt + 0, e1s, e1d, ecnt + 1, deg1, words1);
  conv((float*)x1u, pos1, e1s, e1d, ecnt + 1, deg1, W2, Wr2, b2, y2, N1, C1, C2, mm + 1);

  // ---- pool2 / conv3 ----
  pool(y2, pos1, cnt1, N1, 1728, 30.0f, 8, 512, C2, nid1, cnt2, pos2, x2u,
       e1s, e1d, ecnt + 1, e2s, e2d, ecnt + 2, deg2, words2);
  conv((float*)x2u, pos2, e2s, e2d, ecnt + 2, deg2, W3, Wr3, b3, y3, N2, C2, C3, mm + 2);

  // ---- pool3 / conv4 ----
  pool(y3, pos2, cnt2, N2, 512, 50.0f, 5, 125, C3, nid2, cnt3, pos3, x3u,
       e2s, e2d, ecnt + 2, e3s, e3d, ecnt + 3, deg3, words3);
  conv((float*)x3u, pos3, e3s, e3d, ecnt + 3, deg3, W4, Wr4, b4, y4, N3, C3, C4, mm + 3);

  // ---- MaxPoolingX(80, 64 slots) ----
  k_fill_ordneg<<<nb(256 * 512), THREADS, 0, stream>>>(xgu, 256 * 512);
  k_finalpool<<<nb((long long)N3 * C4), THREADS, 0, stream>>>(y4, pos3, cnt3, xgu, N3 * C4);
  k_xg_fin<<<nb(256 * 512), THREADS, 0, stream>>>(xgu, 256 * 512);

  // ---- FC1 (bandwidth-bound: split-K=64 -> 1024 waves) + FC2 ----
  k_init_bias<<<nb(4 * 1024), THREADS, 0, stream>>>(h, fc1b, 1024, 4 * 1024);
  {
    int kchunks = 64, ntiles = 1024 / 64;
    int waves = ntiles * kchunks;
    k_gemm_splitk_f32wmma<<<dim3((waves + 7) / 8), dim3(THREADS), 0, stream>>>(
        (const float*)xgu, fc1w, h, 4, 64 * 512, 1024, kchunks);
  }
  k_elu<<<nb(4 * 1024), THREADS, 0, stream>>>(h, 4 * 1024);
  k_fc2<<<dim3(4), dim3(THREADS), 0, stream>>>(h, fc2w, fc2b, out);
}